// RecursiveDifferentiationLayer_6176162972381
// MI455X (gfx1250) — compile-verified
//
#include <hip/hip_runtime.h>

#define M_ROWS 1024
#define DFEAT  256
#define LSTR   260          // LDS row stride in floats: 260*4B -> 4-bank step per lane, conflict-free float4
#define BT     32           // block tile (rows on each side)
#define WT     16           // wave tile (WMMA 16x16)

typedef __attribute__((ext_vector_type(2))) float v2f;
typedef __attribute__((ext_vector_type(8))) float v8f;

// ---------------------------------------------------------------------------
// Kernel 1: D[m, :] = X[i[m], :] - X[j[m], :]   and   S[m] = sum_d D[m, d]
// grid = 1024 blocks, block = 256 threads (one thread per feature)
// ---------------------------------------------------------------------------
__global__ __launch_bounds__(DFEAT) void k_pair_diff(
    const float* __restrict__ X, const int* __restrict__ ii, const int* __restrict__ jj,
    float* __restrict__ out, float* __restrict__ rowsum)
{
    __shared__ float red[DFEAT];
    const int m = blockIdx.x;
    const int t = threadIdx.x;
    const int i = ii[m];
    const int j = jj[m];
    const float v = X[(long)i * DFEAT + t] - X[(long)j * DFEAT + t];
    out[(long)m * DFEAT + t] = v;
    red[t] = v;
    __syncthreads();
    #pragma unroll
    for (int s = DFEAT / 2; s > 0; s >>= 1) {
        if (t < s) red[t] += red[t + s];
        __syncthreads();
    }
    if (t == 0) rowsum[m] = red[0];
}

// ---------------------------------------------------------------------------
// Kernel 2: L1 cdist via  D2[a,b] = S[a] + S[b] - 2 * sum_d min(x_ad, x_bd)
// The rank-2 term (S[a] + S[b]) + C is computed EXACTLY by one
// V_WMMA_F32_16X16X4_F32 with C = -2*minsum accumulators (already in the
// WMMA C/D lane layout). block = 128 threads (4 waves, 2x2 of 16x16 tiles).
// DUAL: compile-time second output (D2_1 is stored at two output slots).
// ---------------------------------------------------------------------------
template <bool DUAL>
__global__ __launch_bounds__(128) void k_l1_cdist(
    const float* __restrict__ X, const float* __restrict__ S,
    float* __restrict__ out0, float* __restrict__ out1)
{
    __shared__ float sA[BT][LSTR];
    __shared__ float sB[BT][LSTR];
    __shared__ float sSA[BT];
    __shared__ float sSB[BT];

    const int tid   = threadIdx.x;
    const int nb    = M_ROWS / BT;           // 32 blocks per side
    const int aBase = (blockIdx.x / nb) * BT;
    const int bBase = (blockIdx.x % nb) * BT;

    // Stage 32x256 A/B tiles into LDS (coalesced float4 loads, padded rows).
    for (int idx = tid; idx < BT * (DFEAT / 4); idx += 128) {
        const int r = idx >> 6;              // DFEAT/4 == 64
        const int c = idx & 63;
        const float4 va = ((const float4*)(X + (long)(aBase + r) * DFEAT))[c];
        const float4 vb = ((const float4*)(X + (long)(bBase + r) * DFEAT))[c];
        *(float4*)&sA[r][c * 4] = va;
        *(float4*)&sB[r][c * 4] = vb;
    }
    if (tid < BT) { sSA[tid] = S[aBase + tid]; sSB[tid] = S[bBase + tid]; }
    __syncthreads();

    const int lane = tid & 31;
    const int wave = tid >> 5;
    const int wa   = (wave >> 1) * WT;       // wave tile origin (a side): 0 or 16
    const int wb   = (wave & 1) * WT;        // wave tile origin (b side): 0 or 16
    const int half = lane >> 4;              // 0 -> rows v, 1 -> rows v+8 (WMMA C layout)
    const int n    = lane & 15;              // column (b) within tile

    float acc[8];
    #pragma unroll
    for (int v = 0; v < 8; ++v) acc[v] = 0.f;

    const float* bRow = &sB[wb + n][0];
    #pragma unroll 2
    for (int d = 0; d < DFEAT; d += 4) {
        const float4 xb = *(const float4*)&bRow[d];
        #pragma unroll
        for (int v = 0; v < 8; ++v) {
            const float4 xa = *(const float4*)&sA[wa + v + 8 * half][d];
            acc[v] += fminf(xa.x, xb.x) + fminf(xa.y, xb.y)
                    + fminf(xa.z, xb.z) + fminf(xa.w, xb.w);
        }
    }

    // C = -2 * minsum, in the 16x16 f32 C/D layout (VGPR v: M=v / v+8, N=lane).
    v8f c;
    #pragma unroll
    for (int v = 0; v < 8; ++v) c[v] = -2.0f * acc[v];

    // Branchless fragment construction: all lanes load (broadcast), upper
    // half-wave (K=2,3) zeroed by multiply -> no EXEC-masked LDS loads.
    const float hm = (half == 0) ? 1.0f : 0.0f;
    const float sa = sSA[wa + n];
    const float sb = sSB[wb + n];
    v2f afrag, bfrag;
    afrag[0] = sa * hm;   // K=0 col: S_A[a] (M = lane)   | K=2 -> 0
    afrag[1] = hm;        // K=1 col: ones                | K=3 -> 0
    bfrag[0] = hm;        // K=0 row: ones (N = lane)     | K=2 -> 0
    bfrag[1] = sb * hm;   // K=1 row: S_B[b]              | K=3 -> 0

#if __has_builtin(__builtin_amdgcn_wmma_f32_16x16x4_f32)
    c = __builtin_amdgcn_wmma_f32_16x16x4_f32(
            /*neg_a=*/false, afrag, /*neg_b=*/false, bfrag,
            /*c_mod=*/(short)0, c, /*reuse_a=*/false, /*reuse_b=*/false);
#else
    #pragma unroll
    for (int v = 0; v < 8; ++v)
        c[v] += sSA[wa + v + 8 * half] + sSB[wb + n];
#endif

    // Single base address per output; 8 rows at constant 4KB strides ->
    // global_store_b32 with immediate offsets, half-wave coalesced columns.
    const long base = (long)(aBase + wa + 8 * half) * M_ROWS + (bBase + wb + n);
    float* __restrict__ p0 = out0 + base;
    #pragma unroll
    for (int v = 0; v < 8; ++v) p0[v * M_ROWS] = c[v];
    if constexpr (DUAL) {
        float* __restrict__ p1 = out1 + base;
        #pragma unroll
        for (int v = 0; v < 8; ++v) p1[v * M_ROWS] = c[v];
    }
}

// ---------------------------------------------------------------------------
// Launch: D1/S0 -> D1b/S1 -> cdist(D1)->slot1, cdist(D1b)->slots 0 & 2
// ---------------------------------------------------------------------------
extern "C" void kernel_launch(void* const* d_in, const int* in_sizes, int n_in,
                              void* d_out, int out_size, void* d_ws, size_t ws_size,
                              hipStream_t stream) {
    (void)in_sizes; (void)n_in; (void)out_size; (void)ws_size;

    const float* Z      = (const float*)d_in[0];
    const int*   idx0_i = (const int*)d_in[1];
    const int*   idx0_j = (const int*)d_in[2];
    const int*   idx1_i = (const int*)d_in[3];
    const int*   idx1_j = (const int*)d_in[4];
    float*       out    = (float*)d_out;

    float* D1  = (float*)d_ws;                 // 1024*256
    float* D1b = D1 + (long)M_ROWS * DFEAT;    // 1024*256
    float* S0  = D1b + (long)M_ROWS * DFEAT;   // 1024
    float* S1  = S0 + M_ROWS;                  // 1024

    const long MM = (long)M_ROWS * M_ROWS;

    k_pair_diff<<<M_ROWS, DFEAT, 0, stream>>>(Z,  idx0_i, idx0_j, D1,  S0);
    k_pair_diff<<<M_ROWS, DFEAT, 0, stream>>>(D1, idx1_i, idx1_j, D1b, S1);

    const int nblk = (M_ROWS / BT) * (M_ROWS / BT);   // 1024
    // D2_0 -> output slot 1 (single store)
    k_l1_cdist<false><<<nblk, 128, 0, stream>>>(D1,  S0, out + MM, nullptr);
    // D2_1 -> output slots 0 and 2 (dual store)
    k_l1_cdist<true ><<<nblk, 128, 0, stream>>>(D1b, S1, out, out + 2 * MM);
}